// Type2MoE_6227702579635
// MI455X (gfx1250) — compile-verified
//
#include <hip/hip_runtime.h>
#include <hip/hip_bf16.h>
#include <math.h>

#define HIDDEN 300
#define NEXP 3
#define TOK_PER_BLK 8
#define COLTP 20                    // padded column tiles of 16 (covers 320 >= 300)
#define COLG  (COLTP / 2)           // 10 column groups of 32
#define KQ (HIDDEN / 4)             // 75 K-steps of 4

typedef __attribute__((ext_vector_type(2))) float v2f;
typedef __attribute__((ext_vector_type(8))) float v8f;

// ---------------------------------------------------------------------------
// Kernel 1: gating. One wave (32 lanes) per token. Computes logits (300x3 dot),
// softmax, argmax (first-max tie break), stores idx/gate, and per-block
// expert counts + per-block gate sums (deterministic, no global atomics).
// ---------------------------------------------------------------------------
__global__ void moe_gate(const float* __restrict__ x,
                         const float* __restrict__ wg,
                         int* __restrict__ idx,
                         float* __restrict__ gate,
                         int* __restrict__ bcnt,
                         float* __restrict__ bme,
                         int S)
{
    __shared__ float sh_g[TOK_PER_BLK][NEXP];
    __shared__ int   sh_sel[TOK_PER_BLK];

    const int tid  = threadIdx.x;
    const int wave = tid >> 5;
    const int lane = tid & 31;
    const int t    = blockIdx.x * TOK_PER_BLK + wave;

    if (t < S) {
        float a0 = 0.f, a1 = 0.f, a2 = 0.f;
        const float* xr = x + (size_t)t * HIDDEN;
        for (int k = lane; k < HIDDEN; k += 32) {
            float xv = xr[k];
            a0 += xv * wg[k * NEXP + 0];
            a1 += xv * wg[k * NEXP + 1];
            a2 += xv * wg[k * NEXP + 2];
        }
        for (int off = 16; off > 0; off >>= 1) {
            a0 += __shfl_xor(a0, off, 32);
            a1 += __shfl_xor(a1, off, 32);
            a2 += __shfl_xor(a2, off, 32);
        }
        if (lane == 0) {
            float mx = fmaxf(a0, fmaxf(a1, a2));
            float e0 = expf(a0 - mx), e1 = expf(a1 - mx), e2 = expf(a2 - mx);
            float sm = e0 + e1 + e2;
            float g0 = e0 / sm, g1 = e1 / sm, g2 = e2 / sm;
            int best = 0; float lb = a0;
            if (a1 > lb) { best = 1; lb = a1; }
            if (a2 > lb) { best = 2; lb = a2; }
            idx[t]  = best;
            gate[t] = (best == 0) ? g0 : ((best == 1) ? g1 : g2);
            sh_g[wave][0] = g0; sh_g[wave][1] = g1; sh_g[wave][2] = g2;
            sh_sel[wave] = best;
        }
    } else if (lane == 0) {
        sh_g[wave][0] = sh_g[wave][1] = sh_g[wave][2] = 0.f;
        sh_sel[wave] = -1;
    }
    __syncthreads();

    if (tid < NEXP) {
        float s = 0.f; int c = 0;
        for (int j = 0; j < TOK_PER_BLK; ++j) {
            s += sh_g[j][tid];
            if (sh_sel[j] == tid) c++;
        }
        bme[blockIdx.x * NEXP + tid]  = s;
        bcnt[blockIdx.x * NEXP + tid] = c;
    }
}

// ---------------------------------------------------------------------------
// Kernel 2: single-block scan. Exclusive scan of per-block expert counts
// (gives stable global positions), totals -> valid counts, and l_aux.
// ---------------------------------------------------------------------------
__global__ void moe_scan(const int* __restrict__ bcnt,
                         const float* __restrict__ bme,
                         int* __restrict__ boff,
                         int* __restrict__ valid,
                         float* __restrict__ laux_out,
                         int NB, int S, int CAP)
{
    __shared__ int   totals[NEXP];
    __shared__ float mesum[NEXP];
    const int tid = threadIdx.x;
    if (tid < NEXP) {
        int run = 0; float ms = 0.f;
        for (int b = 0; b < NB; ++b) {
            boff[b * NEXP + tid] = run;
            run += bcnt[b * NEXP + tid];
            ms  += bme[b * NEXP + tid];
        }
        totals[tid] = run;
        mesum[tid]  = ms;
        valid[tid]  = run < CAP ? run : CAP;
    }
    __syncthreads();
    if (tid == 0) {
        float inv = 1.0f / (float)S;
        float l = 0.f;
        for (int e = 0; e < NEXP; ++e)
            l += (mesum[e] * inv) * ((float)totals[e] * inv);
        *laux_out = l * (float)NEXP;
    }
}

// ---------------------------------------------------------------------------
// Kernel 3: stable per-block position assignment + slot table; zero dropped
// output rows. Same block<->token mapping as kernel 1.
// ---------------------------------------------------------------------------
__global__ void moe_pos(const int* __restrict__ idx,
                        const int* __restrict__ boff,
                        int* __restrict__ slot,
                        float* __restrict__ out,
                        int S, int CAP)
{
    __shared__ int sh_drop[TOK_PER_BLK];
    const int tid = threadIdx.x;
    const int t0  = blockIdx.x * TOK_PER_BLK;

    if (tid == 0) {
        int off[NEXP];
        for (int e = 0; e < NEXP; ++e) off[e] = boff[blockIdx.x * NEXP + e];
        for (int j = 0; j < TOK_PER_BLK; ++j) {
            int t = t0 + j;
            int drop = 0;
            if (t < S) {
                int e = idx[t];
                int p = off[e]++;
                if (p < CAP) slot[e * CAP + p] = t;
                else         drop = 1;
            }
            sh_drop[j] = drop;
        }
    }
    __syncthreads();
    for (int j = 0; j < TOK_PER_BLK; ++j) {
        if (sh_drop[j]) {
            float* o = out + (size_t)(t0 + j) * HIDDEN;
            for (int i = tid; i < HIDDEN; i += blockDim.x) o[i] = 0.f;
        }
    }
}

// ---------------------------------------------------------------------------
// Kernel 3b: repack expert weights into WMMA B-fragment order (padded to 20
// column tiles so each wave owns exactly two B streams).
// Layout: wpk[(((e*COLTP + ct)*KQ + kq)*32 + lane)*2 + {0,1}]
//   half = lane>>4, lm = lane&15; ka = 4*kq + 2*half; n = ct*16+lm (clamped)
// ---------------------------------------------------------------------------
__global__ void moe_pack_b(const float* __restrict__ w_exp,
                           float* __restrict__ wpk)
{
    const int gid   = blockIdx.x * blockDim.x + threadIdx.x;
    const int total = NEXP * COLTP * KQ * 32;
    if (gid >= total) return;
    const int lane = gid & 31;
    const int frag = gid >> 5;
    const int kq   = frag % KQ;
    const int rest = frag / KQ;
    const int ct   = rest % COLTP;
    const int e    = rest / COLTP;

    const int half = lane >> 4;
    const int lm   = lane & 15;
    const int ka   = 4 * kq + 2 * half;
    const int n    = ct * 16 + lm;
    const int nc   = (n < HIDDEN) ? n : (HIDDEN - 1);

    const float* wr = w_exp + (size_t)e * HIDDEN * HIDDEN;
    wpk[(size_t)gid * 2 + 0] = wr[(size_t)ka * HIDDEN + nc];
    wpk[(size_t)gid * 2 + 1] = wr[(size_t)(ka + 1) * HIDDEN + nc];
}

// ---------------------------------------------------------------------------
// Kernel 4: per-expert GEMM, WMMA f32 16x16x4 with 2x2 register blocking.
// One wave computes a 32-row x 32-col macro-tile: per K-step it loads
// 2 A fragments + 2 B fragments (4x global_load_b64) and issues 4 WMMAs,
// halving operand traffic vs the 1x1 version. Epilogue adds bias, scales by
// gate, scatter-stores to out[token].
// ---------------------------------------------------------------------------
__global__ void moe_expert_gemm(const float* __restrict__ x,
                                const float* __restrict__ wpk,
                                const float* __restrict__ b_exp,
                                const float* __restrict__ gate,
                                const int* __restrict__ slot,
                                const int* __restrict__ valid,
                                float* __restrict__ out,
                                int CAP)
{
    const int e    = blockIdx.z;
    const int wave = threadIdx.x >> 5;
    const int lane = threadIdx.x & 31;

    const int row_groups = (CAP + 31) / 32;
    const int T = row_groups * COLG;
    const int tile = blockIdx.x * (blockDim.x >> 5) + wave;
    if (tile >= T) return;

    const int rg = tile / COLG;
    const int cg = tile % COLG;
    const int vcnt = valid[e];
    const int rowbase = rg * 32;
    if (rowbase >= vcnt) return;            // wave-uniform

    const int half = lane >> 4;
    const int lm   = lane & 15;

    const int row0 = rowbase + lm;          // rows of upper 16x32 sub-block
    const int row1 = rowbase + 16 + lm;     // rows of lower 16x32 sub-block
    const int tok0 = (row0 < vcnt) ? slot[e * CAP + row0] : 0;
    const int tok1 = (row1 < vcnt) ? slot[e * CAP + row1] : 0;
    const float g0 = gate[tok0];
    const float g1 = gate[tok1];

    const int n0 = cg * 32;
    const int na = n0 + lm;                 // cols of left  32x16 sub-block
    const int nb = n0 + 16 + lm;            // cols of right 32x16 sub-block
    const int nca = (na < HIDDEN) ? na : (HIDDEN - 1);
    const int ncb = (nb < HIDDEN) ? nb : (HIDDEN - 1);

    const float* xr0 = x + (size_t)tok0 * HIDDEN + half * 2;
    const float* xr1 = x + (size_t)tok1 * HIDDEN + half * 2;
    const float* bp0 = wpk + (((size_t)(e * COLTP + 2 * cg) * KQ) * 32 + lane) * 2;
    const float* bp1 = bp0 + (size_t)KQ * 32 * 2;   // next 16-col tile

    v8f c00 = {}, c01 = {}, c10 = {}, c11 = {};
    #pragma unroll 5
    for (int kq = 0; kq < KQ; ++kq) {       // 75 steps: 4 loads + 4 WMMA
        v2f a0 = *(const v2f*)(xr0 + 4 * kq);
        v2f a1 = *(const v2f*)(xr1 + 4 * kq);
        v2f b0 = *(const v2f*)(bp0 + (size_t)kq * 64);
        v2f b1 = *(const v2f*)(bp1 + (size_t)kq * 64);
        c00 = __builtin_amdgcn_wmma_f32_16x16x4_f32(false, a0, false, b0, (short)0, c00, false, false);
        c01 = __builtin_amdgcn_wmma_f32_16x16x4_f32(false, a0, false, b1, (short)0, c01, false, false);
        c10 = __builtin_amdgcn_wmma_f32_16x16x4_f32(false, a1, false, b0, (short)0, c10, false, false);
        c11 = __builtin_amdgcn_wmma_f32_16x16x4_f32(false, a1, false, b1, (short)0, c11, false, false);
    }

    const float bias_a = b_exp[e * HIDDEN + nca];
    const float bias_b = b_exp[e * HIDDEN + ncb];

    union { v8f v; float f[8]; } u00, u01, u10, u11;
    u00.v = c00; u01.v = c01; u10.v = c10; u11.v = c11;

    #pragma unroll
    for (int r = 0; r < 8; ++r) {
        const int m = r + 8 * half;          // C layout: M = r + 8*(lane>=16)
        // upper rows
        if (rowbase + m < vcnt) {
            const int   tm = __shfl(tok0, m, 32);
            const float gm = __shfl(g0,   m, 32);
            if (na < HIDDEN) out[(size_t)tm * HIDDEN + na] = (u00.f[r] + bias_a) * gm;
            if (nb < HIDDEN) out[(size_t)tm * HIDDEN + nb] = (u01.f[r] + bias_b) * gm;
        }
        // lower rows
        if (rowbase + 16 + m < vcnt) {
            const int   tm = __shfl(tok1, m, 32);
            const float gm = __shfl(g1,   m, 32);
            if (na < HIDDEN) out[(size_t)tm * HIDDEN + na] = (u10.f[r] + bias_a) * gm;
            if (nb < HIDDEN) out[(size_t)tm * HIDDEN + nb] = (u11.f[r] + bias_b) * gm;
        }
    }
}

// ---------------------------------------------------------------------------
extern "C" void kernel_launch(void* const* d_in, const int* in_sizes, int n_in,
                              void* d_out, int out_size, void* d_ws, size_t ws_size,
                              hipStream_t stream)
{
    (void)n_in; (void)out_size; (void)ws_size;

    const float* x     = (const float*)d_in[0];   // [8,8192,300]
    const float* wg    = (const float*)d_in[1];   // [300,3]
    const float* w_exp = (const float*)d_in[2];   // [3,300,300]
    const float* b_exp = (const float*)d_in[3];   // [3,300]
    float* out = (float*)d_out;                   // [S*300] ++ [l_aux]

    const int S   = in_sizes[0] / HIDDEN;                 // 65536 tokens
    const int CAP = (S + NEXP - 1) / NEXP;                // ceil(S/E), cap_factor=1
    const int NB  = (S + TOK_PER_BLK - 1) / TOK_PER_BLK;  // gating blocks

    // workspace carve (16B aligned chunks)
    char* p = (char*)d_ws;
    auto carve = [&](size_t bytes) { char* q = p; p += (bytes + 15) & ~(size_t)15; return q; };
    int*   idx   = (int*)  carve((size_t)S * sizeof(int));
    float* gate  = (float*)carve((size_t)S * sizeof(float));
    int*   bcnt  = (int*)  carve((size_t)NB * NEXP * sizeof(int));
    float* bme   = (float*)carve((size_t)NB * NEXP * sizeof(float));
    int*   boff  = (int*)  carve((size_t)NB * NEXP * sizeof(int));
    int*   valid = (int*)  carve(NEXP * sizeof(int));
    int*   slot  = (int*)  carve((size_t)NEXP * CAP * sizeof(int));
    float* wpk   = (float*)carve((size_t)NEXP * COLTP * KQ * 32 * 2 * sizeof(float));

    // B repack is independent of gating -> launch first
    {
        const int total = NEXP * COLTP * KQ * 32;
        moe_pack_b<<<(total + 255) / 256, 256, 0, stream>>>(w_exp, wpk);
    }

    moe_gate<<<NB, TOK_PER_BLK * 32, 0, stream>>>(x, wg, idx, gate, bcnt, bme, S);
    moe_scan<<<1, 32, 0, stream>>>(bcnt, bme, boff, valid,
                                   out + (size_t)S * HIDDEN, NB, S, CAP);
    moe_pos<<<NB, 256, 0, stream>>>(idx, boff, slot, out, S, CAP);

    const int row_groups = (CAP + 31) / 32;
    const int T = row_groups * COLG;              // macro-tiles per expert
    dim3 grid((T + 7) / 8, 1, NEXP);              // 8 waves (macro-tiles) / block
    moe_expert_gemm<<<grid, 256, 0, stream>>>(x, wpk, b_exp, gate,
                                              slot, valid, out, CAP);
}